// L2U_Mamba_74474732912933
// MI455X (gfx1250) — compile-verified
//
#include <hip/hip_runtime.h>
#include <cmath>
#include <stdint.h>

typedef float v2f __attribute__((ext_vector_type(2)));
typedef float v8f __attribute__((ext_vector_type(8)));
typedef uint32_t u32x4 __attribute__((ext_vector_type(4)));
typedef uint32_t u32x8 __attribute__((ext_vector_type(8)));

#define P_TOT      65536
#define DI         128     // D_INNER
#define DS         16      // D_STATE
#define CONV_CHUNK 512
#define NBIG       160     // 128 dt + 16 B + 16 C
#define SCL        256     // scan chunk length
#define NSC        256     // number of scan chunks (SCL*NSC == P_TOT)

// ---- workspace layout (float offsets) ----
constexpr size_t W_XZ   = 0;                       // 256 x 2 folded in_proj∘mamba_in
constexpr size_t B_XZ   = 512;                     // 256
constexpr size_t W_BIG  = 768;                     // 160 x 128 folded [dtproj∘xproj ; xprojB ; xprojC]
constexpr size_t B_BIG  = 21248;                   // 160
constexpr size_t A_NEG  = 21408;                   // 128 x 16 = -exp(A_log)
constexpr size_t W_OUT  = 23456;                   // 16 x 128 folded outproj∘mamba_out (rows 3..15 = 0)
constexpr size_t B_OUT  = 25504;                   // 16
constexpr size_t XCONV  = 32768;                   // P x 128
constexpr size_t ZSILU  = XCONV  + (size_t)P_TOT * DI;
constexpr size_t DT_    = ZSILU  + (size_t)P_TOT * DI;
constexpr size_t BSEL   = DT_    + (size_t)P_TOT * DI;
constexpr size_t CSEL   = BSEL   + (size_t)P_TOT * DS;
constexpr size_t CARRYA = CSEL   + (size_t)P_TOT * DS;
constexpr size_t CARRYB = CARRYA + (size_t)NSC * 2048;
constexpr size_t YGATED = CARRYB + (size_t)NSC * 2048;

__device__ __forceinline__ float sigmoidf_(float x){
    return __builtin_amdgcn_rcpf(1.0f + __expf(-x));
}
__device__ __forceinline__ float siluf_(float x){ return x * sigmoidf_(x); }
__device__ __forceinline__ float softplusf_(float x){
    return x > 20.0f ? x : __logf(1.0f + __expf(x));
}
__device__ __forceinline__ float fast_tanhf_(float x){
    float e = __expf(2.0f * x);
    return (e - 1.0f) * __builtin_amdgcn_rcpf(e + 1.0f);
}

// ---- Tensor Data Mover: contiguous 1D global->LDS copy of ndwords DWORDs ----
// D# group0/group1 built per CDNA5 ISA sec 8.3/8.4. Wave-level op; caller must
// guard so only one wave per block issues it, then s_wait_tensorcnt + barrier.
__device__ __forceinline__ void tdm_load_1d(uint32_t lds_off, const void* gptr, uint32_t ndwords)
{
    uint64_t ga = (uint64_t)(uintptr_t)gptr;
    u32x4 g0;
    g0[0] = 1u;                                                   // count=1, user mode
    g0[1] = lds_off;                                              // lds_addr (bytes)
    g0[2] = (uint32_t)ga;                                         // global_addr[31:0]
    g0[3] = ((uint32_t)(ga >> 32) & 0x01FFFFFFu) | (2u << 30);    // addr[56:32] | type=2
    u32x8 g1;
    g1[0] = 2u << 16;                                             // data_size=4B, no mask/pad/iter
    g1[1] = (ndwords & 0xFFFFu) << 16;                            // tensor_dim0[15:0]
    g1[2] = (ndwords >> 16) | (1u << 16);                         // tensor_dim0[31:16] | tensor_dim1=1
    g1[3] = (ndwords & 0xFFFFu) << 16;                            // tile_dim0 = ndwords
    g1[4] = 1u;                                                   // tile_dim1=1, tile_dim2=0
    g1[5] = ndwords;                                              // tensor_dim0_stride[31:0]
    g1[6] = 0u;
    g1[7] = 0u;
    asm volatile("tensor_load_to_lds %0, %1" :: "s"(g0), "s"(g1) : "memory");
}
__device__ __forceinline__ void tdm_wait0()
{
    asm volatile("s_wait_tensorcnt 0x0" ::: "memory");
}

// ---------------- Kernel 0: fold all weights on device ----------------
__global__ void k0_fold(const float* __restrict__ in_proj_w, const float* __restrict__ in_proj_b,
                        const float* __restrict__ mamba_in_w, const float* __restrict__ mamba_in_b,
                        const float* __restrict__ xproj_w, const float* __restrict__ xproj_b,
                        const float* __restrict__ dtproj_w, const float* __restrict__ dtproj_b,
                        const float* __restrict__ A_log,
                        const float* __restrict__ mamba_out_w, const float* __restrict__ mamba_out_b,
                        const float* __restrict__ outproj_w, const float* __restrict__ outproj_b,
                        float* __restrict__ ws)
{
    int t = threadIdx.x;   // 256 threads
    {
        int o = t;
        float a0 = 0.f, a1 = 0.f, ab = mamba_in_b[o];
        for (int m = 0; m < 64; ++m) {
            float w = mamba_in_w[o * 64 + m];
            a0 += w * in_proj_w[m * 2 + 0];
            a1 += w * in_proj_w[m * 2 + 1];
            ab += w * in_proj_b[m];
        }
        ws[W_XZ + o * 2 + 0] = a0;
        ws[W_XZ + o * 2 + 1] = a1;
        ws[B_XZ + o] = ab;
    }
    for (int idx = t; idx < NBIG * DI; idx += 256) {
        int r = idx >> 7, k = idx & 127;
        float v;
        if (r < 128) {
            v = 0.f;
            for (int j = 0; j < 4; ++j) v += dtproj_w[r * 4 + j] * xproj_w[j * 128 + k];
        } else if (r < 144) {
            v = xproj_w[(4 + (r - 128)) * 128 + k];
        } else {
            v = xproj_w[(20 + (r - 144)) * 128 + k];
        }
        ws[W_BIG + idx] = v;
    }
    for (int r = t; r < NBIG; r += 256) {
        float v;
        if (r < 128) {
            v = dtproj_b[r];
            for (int j = 0; j < 4; ++j) v += dtproj_w[r * 4 + j] * xproj_b[j];
        } else if (r < 144) v = xproj_b[4 + (r - 128)];
        else                v = xproj_b[20 + (r - 144)];
        ws[B_BIG + r] = v;
    }
    for (int idx = t; idx < DI * DS; idx += 256) ws[A_NEG + idx] = -__expf(A_log[idx]);
    for (int idx = t; idx < 16 * DI; idx += 256) {
        int r = idx >> 7, k = idx & 127;
        float v = 0.f;
        if (r < 3) for (int m = 0; m < 64; ++m) v += outproj_w[r * 64 + m] * mamba_out_w[m * 128 + k];
        ws[W_OUT + idx] = v;
    }
    if (t < 16) {
        float v = 0.f;
        if (t < 3) {
            v = outproj_b[t];
            for (int m = 0; m < 64; ++m) v += outproj_w[t * 64 + m] * mamba_out_b[m];
        }
        ws[B_OUT + t] = v;
    }
}

// ---------------- Kernel 1: folded in-proj + depthwise conv + silu ----------------
__global__ void k1_frontend(const float* __restrict__ inp,
                            const float* __restrict__ Wxz, const float* __restrict__ bxz,
                            const float* __restrict__ conv_w, const float* __restrict__ conv_b,
                            float* __restrict__ xconv, float* __restrict__ zsilu)
{
    int d = threadIdx.x;                 // 128 channels
    int p0 = blockIdx.x * CONV_CHUNK;    // 128 conv chunks
    float wx0 = Wxz[d * 2], wx1 = Wxz[d * 2 + 1], bx = bxz[d];
    float wz0 = Wxz[(d + 128) * 2], wz1 = Wxz[(d + 128) * 2 + 1], bz = bxz[d + 128];
    float cw0 = conv_w[d * 4 + 0], cw1 = conv_w[d * 4 + 1];
    float cw2 = conv_w[d * 4 + 2], cw3 = conv_w[d * 4 + 3];
    float cb = conv_b[d];
    float h1 = 0.f, h2 = 0.f, h3 = 0.f;  // causal history, zero-padded at chunk start
    for (int i = 0; i < CONV_CHUNK; ++i) {
        int p = p0 + i;
        float in0 = inp[p * 2 + 0], in1 = inp[p * 2 + 1];   // wave-uniform -> scalar loads
        float xv = in0 * wx0 + in1 * wx1 + bx;
        float zv = in0 * wz0 + in1 * wz1 + bz;
        float cacc = cb + cw3 * xv + cw2 * h1 + cw1 * h2 + cw0 * h3;
        h3 = h2; h2 = h1; h1 = xv;
        xconv[(size_t)p * DI + d] = siluf_(cacc);
        zsilu[(size_t)p * DI + d] = siluf_(zv);
    }
}

// ---------------- Kernel 2: WMMA f32 GEMM  x_conv(65536x128) @ W_big.T(128x160) ----------------
__global__ void k2_gemm_xdbl(const float* __restrict__ xconv,
                             const float* __restrict__ Wbig,
                             const float* __restrict__ bbig,
                             float* __restrict__ dt_out,
                             float* __restrict__ Bsel,
                             float* __restrict__ Csel)
{
    int lane = threadIdx.x & 31;
    int wid  = threadIdx.x >> 5;
    int mtile = blockIdx.x * 8 + wid;
    int row0 = mtile * 16;
    int ml = lane & 15;      // M (for A) / N (for B,C,D)
    int half = lane >> 4;    // K sub-group / M+8 for C,D

    v8f acc[10];
#pragma unroll
    for (int nt = 0; nt < 10; ++nt) {
        float bv = bbig[nt * 16 + ml];
#pragma unroll
        for (int r = 0; r < 8; ++r) acc[nt][r] = bv;
    }

    const float* arow = xconv + (size_t)(row0 + ml) * DI + 2 * half;
#pragma unroll 1
    for (int kb = 0; kb < DI; kb += 4) {
        v2f a = *(const v2f*)(arow + kb);                       // A 16x4 fragment
#pragma unroll
        for (int nt = 0; nt < 10; ++nt) {
            const float* bp = Wbig + (size_t)(nt * 16 + ml) * DI + kb + 2 * half;
            v2f b = *(const v2f*)bp;                            // B 4x16 fragment
            acc[nt] = __builtin_amdgcn_wmma_f32_16x16x4_f32(
                false, a, false, b, (short)0, acc[nt], false, false);
        }
    }

#pragma unroll
    for (int nt = 0; nt < 10; ++nt) {
        int j = nt * 16 + ml;
#pragma unroll
        for (int r = 0; r < 8; ++r) {
            int row = row0 + r + 8 * half;
            float v = acc[nt][r];
            if (j < 128)      dt_out[(size_t)row * DI + j] = softplusf_(v);
            else if (j < 144) Bsel[(size_t)row * DS + (j - 128)] = v;
            else              Csel[(size_t)row * DS + (j - 144)] = v;
        }
    }
}

// ---------------- Kernel 3a: per-chunk scan carries ----------------
__global__ void k3a_carries(const float* __restrict__ dt_, const float* __restrict__ xconv,
                            const float* __restrict__ Bsel, const float* __restrict__ Aneg,
                            float* __restrict__ carryA, float* __restrict__ carryB)
{
    __shared__ float sB[SCL * DS];
    int d = threadIdx.x;
    int c = blockIdx.x;
    int p0 = c * SCL;
    // TDM: one wave DMA-copies the 16KB B tile into LDS, others wait at barrier
    if ((threadIdx.x >> 5) == 0) {
        tdm_load_1d((uint32_t)(size_t)sB, Bsel + (size_t)p0 * DS, SCL * DS);
        tdm_wait0();
    }
    float Ar[16];
#pragma unroll
    for (int s = 0; s < 16; ++s) Ar[s] = Aneg[d * 16 + s];
    __syncthreads();
    float a[16], b[16];
#pragma unroll
    for (int s = 0; s < 16; ++s) { a[s] = 1.0f; b[s] = 0.0f; }
    for (int i = 0; i < SCL; ++i) {
        int p = p0 + i;
        float dtv = dt_[(size_t)p * DI + d];
        float u   = xconv[(size_t)p * DI + d];
        __builtin_prefetch(&dt_[(size_t)(p + 16) * DI + d], 0, 0);
        float du = dtv * u;
#pragma unroll
        for (int s = 0; s < 16; ++s) {
            float e = __expf(dtv * Ar[s]);
            a[s] *= e;
            b[s] = b[s] * e + du * sB[i * 16 + s];
        }
    }
    size_t base = (size_t)c * 2048 + (size_t)d * 16;
#pragma unroll
    for (int s = 0; s < 16; ++s) { carryA[base + s] = a[s]; carryB[base + s] = b[s]; }
}

// ---------------- Kernel 3b: scan over chunk carries (2048 channels) ----------------
__global__ void k3b_chunkscan(float* __restrict__ carryA, float* __restrict__ carryB)
{
    int ch = blockIdx.x * blockDim.x + threadIdx.x;  // 0..2047, coalesced per step
    float pa = carryA[ch], pb = carryB[ch];
    for (int i = 1; i < NSC; ++i) {
        size_t idx = (size_t)i * 2048 + ch;
        float a = carryA[idx], b = carryB[idx];
        pb = pb * a + b;
        pa = pa * a;
        carryA[idx] = pa;
        carryB[idx] = pb;
    }
}

// ---------------- Kernel 3c: replay with carry-in, fuse contraction + gating ----------------
__global__ void k3c_scan_apply(const float* __restrict__ dt_, const float* __restrict__ xconv,
                               const float* __restrict__ zsilu,
                               const float* __restrict__ Bsel, const float* __restrict__ Csel,
                               const float* __restrict__ Aneg, const float* __restrict__ Dp,
                               const float* __restrict__ carryB,
                               float* __restrict__ ygated)
{
    __shared__ float sB[SCL * DS];
    __shared__ float sC[SCL * DS];
    int d = threadIdx.x;
    int c = blockIdx.x;
    int p0 = c * SCL;
    // TDM: two descriptors issued back-to-back (in-order per wave), single wait
    if ((threadIdx.x >> 5) == 0) {
        tdm_load_1d((uint32_t)(size_t)sB, Bsel + (size_t)p0 * DS, SCL * DS);
        tdm_load_1d((uint32_t)(size_t)sC, Csel + (size_t)p0 * DS, SCL * DS);
        tdm_wait0();
    }
    float Ar[16];
#pragma unroll
    for (int s = 0; s < 16; ++s) Ar[s] = Aneg[d * 16 + s];
    float h[16];
    if (c == 0) {
#pragma unroll
        for (int s = 0; s < 16; ++s) h[s] = 0.0f;
    } else {
        size_t base = (size_t)(c - 1) * 2048 + (size_t)d * 16;
#pragma unroll
        for (int s = 0; s < 16; ++s) h[s] = carryB[base + s];
    }
    float Dv = Dp[d];
    __syncthreads();
    for (int i = 0; i < SCL; ++i) {
        int p = p0 + i;
        float dtv = dt_[(size_t)p * DI + d];
        float u   = xconv[(size_t)p * DI + d];
        float zs  = zsilu[(size_t)p * DI + d];
        __builtin_prefetch(&xconv[(size_t)(p + 16) * DI + d], 0, 0);
        float du = dtv * u;
        float y = 0.0f;
#pragma unroll
        for (int s = 0; s < 16; ++s) {
            float e = __expf(dtv * Ar[s]);
            h[s] = h[s] * e + du * sB[i * 16 + s];
            y += h[s] * sC[i * 16 + s];
        }
        y = (y + Dv * u) * zs;
        ygated[(size_t)p * DI + d] = y;
    }
}

// ---------------- Kernel 4: WMMA f32 folded output projection + tanh gate ----------------
__global__ void k4_outproj(const float* __restrict__ ygated,
                           const float* __restrict__ Wout,
                           const float* __restrict__ bout,
                           float* __restrict__ out)
{
    int lane = threadIdx.x & 31;
    int wid  = threadIdx.x >> 5;
    int mtile = blockIdx.x * 8 + wid;
    int row0 = mtile * 16;
    int ml = lane & 15;
    int half = lane >> 4;

    v8f acc;
    float bv = bout[ml];
#pragma unroll
    for (int r = 0; r < 8; ++r) acc[r] = bv;

    const float* arow = ygated + (size_t)(row0 + ml) * DI + 2 * half;
    const float* brow = Wout   + (size_t)ml * DI + 2 * half;
#pragma unroll 1
    for (int kb = 0; kb < DI; kb += 4) {
        v2f a = *(const v2f*)(arow + kb);
        v2f b = *(const v2f*)(brow + kb);
        acc = __builtin_amdgcn_wmma_f32_16x16x4_f32(
            false, a, false, b, (short)0, acc, false, false);
    }

    int base = half * 16;
#pragma unroll
    for (int r = 0; r < 8; ++r) {
        float f0 = __shfl(acc[r], base + 0, 32);   // ifc[:,0]
        float f2 = __shfl(acc[r], base + 2, 32);   // ifc[:,2]
        float it = fast_tanhf_(f0) * f0;
        float gt = fast_tanhf_(f2);
        if (ml == 0) out[row0 + r + 8 * half] = it * gt;
    }
}

extern "C" void kernel_launch(void* const* d_in, const int* in_sizes, int n_in,
                              void* d_out, int out_size, void* d_ws, size_t ws_size,
                              hipStream_t stream)
{
    const float* inp         = (const float*)d_in[0];
    const float* in_proj_w   = (const float*)d_in[1];
    const float* in_proj_b   = (const float*)d_in[2];
    const float* mamba_in_w  = (const float*)d_in[3];
    const float* mamba_in_b  = (const float*)d_in[4];
    const float* conv_w      = (const float*)d_in[5];
    const float* conv_b      = (const float*)d_in[6];
    const float* xproj_w     = (const float*)d_in[7];
    const float* xproj_b     = (const float*)d_in[8];
    const float* dtproj_w    = (const float*)d_in[9];
    const float* dtproj_b    = (const float*)d_in[10];
    const float* A_log       = (const float*)d_in[11];
    const float* D_param     = (const float*)d_in[12];
    const float* mamba_out_w = (const float*)d_in[13];
    const float* mamba_out_b = (const float*)d_in[14];
    const float* outproj_w   = (const float*)d_in[15];
    const float* outproj_b   = (const float*)d_in[16];
    float* ws  = (float*)d_ws;
    float* out = (float*)d_out;

    k0_fold<<<1, 256, 0, stream>>>(in_proj_w, in_proj_b, mamba_in_w, mamba_in_b,
                                   xproj_w, xproj_b, dtproj_w, dtproj_b, A_log,
                                   mamba_out_w, mamba_out_b, outproj_w, outproj_b, ws);
    k1_frontend<<<128, 128, 0, stream>>>(inp, ws + W_XZ, ws + B_XZ, conv_w, conv_b,
                                         ws + XCONV, ws + ZSILU);
    k2_gemm_xdbl<<<512, 256, 0, stream>>>(ws + XCONV, ws + W_BIG, ws + B_BIG,
                                          ws + DT_, ws + BSEL, ws + CSEL);
    k3a_carries<<<256, 128, 0, stream>>>(ws + DT_, ws + XCONV, ws + BSEL, ws + A_NEG,
                                         ws + CARRYA, ws + CARRYB);
    k3b_chunkscan<<<8, 256, 0, stream>>>(ws + CARRYA, ws + CARRYB);
    k3c_scan_apply<<<256, 128, 0, stream>>>(ws + DT_, ws + XCONV, ws + ZSILU,
                                            ws + BSEL, ws + CSEL, ws + A_NEG, D_param,
                                            ws + CARRYB, ws + YGATED);
    k4_outproj<<<512, 256, 0, stream>>>(ws + YGATED, ws + W_OUT, ws + B_OUT, out);
}